// NBInjector_48636209660030
// MI455X (gfx1250) — compile-verified
//
#include <hip/hip_runtime.h>
#include <math.h>

// ---------------------------------------------------------------------------
// MI455X / gfx1250 implementation.
//   k1: fp32 WMMA sims GEMM (2048x768 x 768x50000) + per-chunk partial top-3
//       (dual accumulator chains, branchless top-3)
//   k2: merge partial top-3, apply 1/||v|| scaling, gather nb_vecs, build
//       padded "combined" rows (1671 -> 1680, zero padded)
//   k3: combined @ W1 + b1 -> exact (erf) GELU        [fp32 WMMA]
//   k4: h @ W2 + b2                                   [fp32 WMMA]
//   k5: layernorm (eps=1e-5, no affine)
// ---------------------------------------------------------------------------

#define CVD 768
#define VD 50000
#define DNBD 300
#define TOTAL_IN 1671
#define CT 1680          // combined width padded to multiple of 16
#define ROWS 2048        // B*P

#define K1_BLOCK_M 64
#define K1_THREADS 128
#define NCHUNK 25
#define TILES_PER_CHUNK 125   // 25 * 125 * 16 = 50000
#define APAD 772              // padded LDS row stride (conflict-free)

typedef __attribute__((ext_vector_type(2))) float v2f;
typedef __attribute__((ext_vector_type(8))) float v8f;

__device__ __forceinline__ v8f wmma_f32_4(v2f a, v2f b, v8f c) {
  // 8 args: (neg_a, A, neg_b, B, c_mod, C, reuse_a, reuse_b)
  return __builtin_amdgcn_wmma_f32_16x16x4_f32(false, a, false, b, (short)0, c,
                                               false, false);
}

// Branchless sorted-top3 insertion: lowers to v_cndmask chains, no exec-mask
// divergence (the branchy version cost v_cmpx + s_cbranch + exec restores).
__device__ __forceinline__ void top3_insert(float s, int c,
                                            float &v0, float &v1, float &v2,
                                            int &i0, int &i1, int &i2) {
  const bool b0 = s > v0, b1 = s > v1, b2 = s > v2;
  v2 = b1 ? v1 : (b2 ? s : v2);
  i2 = b1 ? i1 : (b2 ? c : i2);
  v1 = b0 ? v0 : (b1 ? s : v1);
  i1 = b0 ? i0 : (b1 ? c : i1);
  v0 = b0 ? s : v0;
  i0 = b0 ? c : i0;
}

// -------------------------- kernel 1: sims + partial top3 ------------------
__global__ __launch_bounds__(K1_THREADS)
void k1_sims_topk(const float* __restrict__ vseq, const float* __restrict__ Tc,
                  float* __restrict__ pval, int* __restrict__ pidx) {
  extern __shared__ float smem[];
  float* ldsA = smem;                       // 64 x APAD
  float* ldsB = smem + K1_BLOCK_M * APAD;   // 16 x APAD
  const int tid  = threadIdx.x;
  const int wave = tid >> 5, lane = tid & 31;
  const int l16  = lane & 15;
  const int kk   = (lane >> 4) << 1;        // K sub-offset per WMMA A/B layout
  const int rowbase_wg = blockIdx.x * K1_BLOCK_M;
  const int chunk = blockIdx.y;

  // Stage the 64x768 query slab once (fits thanks to 320KB LDS per WGP).
  for (int i = tid; i < K1_BLOCK_M * (CVD / 4); i += K1_THREADS) {
    int r = i / (CVD / 4), c4 = i % (CVD / 4);
    *(float4*)&ldsA[r * APAD + 4 * c4] =
        *(const float4*)&vseq[(size_t)(rowbase_wg + r) * CVD + 4 * c4];
  }

  float tv[8][3]; int ti[8][3];
  #pragma unroll
  for (int r = 0; r < 8; ++r) {
    tv[r][0] = tv[r][1] = tv[r][2] = -INFINITY;
    ti[r][0] = ti[r][1] = ti[r][2] = 0;
  }

  const int aOff = (wave * 16 + l16) * APAD + kk;
  const int bOff = l16 * APAD + kk;

  __syncthreads();

  for (int nt = 0; nt < TILES_PER_CHUNK; ++nt) {
    const int v0 = (chunk * TILES_PER_CHUNK + nt) * 16;
    __syncthreads();   // protect ldsB from previous iteration's readers
    for (int i = tid; i < 16 * (CVD / 4); i += K1_THREADS) {
      int n = i / (CVD / 4), c4 = i % (CVD / 4);
      *(float4*)&ldsB[n * APAD + 4 * c4] =
          *(const float4*)&Tc[(size_t)(v0 + n) * CVD + 4 * c4];
    }
    __syncthreads();

    if (nt + 1 < TILES_PER_CHUNK) {   // speculative prefetch of next bank tile
      __builtin_prefetch(&Tc[(size_t)(v0 + 16 + l16) * CVD + (size_t)wave * 192], 0, 1);
    }

    // Two independent accumulation chains so consecutive WMMAs are not
    // serialized on the same D registers.
    v8f acc0 = {}, acc1 = {};
    #pragma unroll 4
    for (int k0 = 0; k0 < CVD; k0 += 8) {
      v2f a0 = *(const v2f*)&ldsA[aOff + k0];
      v2f b0 = *(const v2f*)&ldsB[bOff + k0];
      acc0 = wmma_f32_4(a0, b0, acc0);
      v2f a1 = *(const v2f*)&ldsA[aOff + k0 + 4];
      v2f b1 = *(const v2f*)&ldsB[bOff + k0 + 4];
      acc1 = wmma_f32_4(a1, b1, acc1);
    }
    v8f acc = acc0 + acc1;

    const int col = v0 + l16;
    #pragma unroll
    for (int r = 0; r < 8; ++r)
      top3_insert(acc[r], col, tv[r][0], tv[r][1], tv[r][2],
                  ti[r][0], ti[r][1], ti[r][2]);
  }

  // Merge the 16 per-lane top3 sets of each row (reuse ldsA region).
  __syncthreads();
  float* MV = smem + wave * 1536;
  int*   MI = (int*)smem + wave * 1536 + 768;
  #pragma unroll
  for (int r = 0; r < 8; ++r) {
    int row_local = r + ((lane >> 4) << 3);
    int base = (row_local * 16 + l16) * 3;
    #pragma unroll
    for (int j = 0; j < 3; ++j) { MV[base + j] = tv[r][j]; MI[base + j] = ti[r][j]; }
  }
  __syncthreads();
  if (lane < 16) {
    float w0 = -INFINITY, w1 = -INFINITY, w2 = -INFINITY;
    int j0 = 0, j1 = 0, j2 = 0;
    const int rb = lane * 48;
    for (int t = 0; t < 48; ++t)
      top3_insert(MV[rb + t], MI[rb + t], w0, w1, w2, j0, j1, j2);
    const int row = rowbase_wg + wave * 16 + lane;
    const size_t o = ((size_t)chunk * ROWS + row) * 3;
    pval[o + 0] = w0; pval[o + 1] = w1; pval[o + 2] = w2;
    pidx[o + 0] = j0; pidx[o + 1] = j1; pidx[o + 2] = j2;
  }
}

// ----------------- kernel 2: merge chunks, norms, build combined -----------
__global__ __launch_bounds__(128)
void k2_merge_build(const float* __restrict__ vseq, const float* __restrict__ nb,
                    const float* __restrict__ pval, const int* __restrict__ pidx,
                    float* __restrict__ combined) {
  const int wave = threadIdx.x >> 5, lane = threadIdx.x & 31;
  const int row = blockIdx.x * 4 + wave;

  // 1/max(||v||, eps): raw dots were computed unscaled (bank is unit-norm).
  float s = 0.f;
  for (int c = lane; c < CVD; c += 32) {
    float x = vseq[(size_t)row * CVD + c];
    s += x * x;
  }
  #pragma unroll
  for (int off = 16; off; off >>= 1) s += __shfl_xor(s, off, 32);
  const float inv = 1.f / fmaxf(sqrtf(s), 1e-12f);

  float w0 = -INFINITY, w1 = -INFINITY, w2 = -INFINITY;
  int j0 = 0, j1 = 0, j2 = 0;
  if (lane == 0) {
    for (int ch = 0; ch < NCHUNK; ++ch) {
      const size_t o = ((size_t)ch * ROWS + row) * 3;
      for (int j = 0; j < 3; ++j)
        top3_insert(pval[o + j], pidx[o + j], w0, w1, w2, j0, j1, j2);
    }
    w0 *= inv; w1 *= inv; w2 *= inv;
  }
  w0 = __shfl(w0, 0, 32); w1 = __shfl(w1, 0, 32); w2 = __shfl(w2, 0, 32);
  j0 = __shfl(j0, 0, 32); j1 = __shfl(j1, 0, 32); j2 = __shfl(j2, 0, 32);

  float* out = combined + (size_t)row * CT;
  for (int c = lane; c < CT; c += 32) {
    float x;
    if (c < CVD) {
      x = vseq[(size_t)row * CVD + c];
    } else if (c < CVD + 3 * DNBD) {
      int q = c - CVD;
      int k = q / DNBD, d = q - k * DNBD;
      int id = (k == 0) ? j0 : ((k == 1) ? j1 : j2);
      x = nb[(size_t)id * DNBD + d];
    } else if (c < TOTAL_IN) {
      int k = c - (CVD + 3 * DNBD);
      x = (k == 0) ? w0 : ((k == 1) ? w1 : w2);
    } else {
      x = 0.f;   // zero pad 1671..1679
    }
    out[c] = x;
  }
}

// ------------------- kernels 3/4: fp32 WMMA GEMM + epilogue ----------------
// 64x64 output tile per WG: 4 waves x (16 rows x 4 column tiles).
template <int KP, int STRIDE, int KTOTAL, int KVALID, int ACT>
__global__ __launch_bounds__(128)
void gemm_kernel(const float* __restrict__ A, int lda,
                 const float* __restrict__ Bm,     // (KVALID x 768) row-major
                 const float* __restrict__ bias,
                 float* __restrict__ C) {          // (ROWS x 768)
  extern __shared__ float smem[];
  float* lA = smem;                 // 64 x STRIDE
  float* lB = smem + 64 * STRIDE;   // 64 x STRIDE (col-major panel of Bm)
  const int tid = threadIdx.x, wave = tid >> 5, lane = tid & 31;
  const int l16 = lane & 15, kk = (lane >> 4) << 1;
  const int rowbase = blockIdx.x * 64, colbase = blockIdx.y * 64;

  v8f acc[4] = {};
  const int aOff = (wave * 16 + l16) * STRIDE + kk;

  for (int kp = 0; kp < KTOTAL; kp += KP) {
    __syncthreads();
    for (int i = tid; i < 64 * KP; i += 128) {
      int r = i / KP, k = i % KP;
      lA[r * STRIDE + k] = A[(size_t)(rowbase + r) * lda + kp + k];
    }
    for (int i = tid; i < 64 * KP; i += 128) {
      int k = i >> 6, n = i & 63;
      int kg = kp + k;
      lB[n * STRIDE + k] = (kg < KVALID) ? Bm[(size_t)kg * 768 + colbase + n] : 0.f;
    }
    __syncthreads();
    #pragma unroll 4
    for (int k0 = 0; k0 < KP; k0 += 4) {
      v2f a = *(const v2f*)&lA[aOff + k0];
      #pragma unroll
      for (int j = 0; j < 4; ++j) {   // 4 independent accumulation chains
        v2f b = *(const v2f*)&lB[(l16 + 16 * j) * STRIDE + kk + k0];
        acc[j] = wmma_f32_4(a, b, acc[j]);
      }
    }
  }

  #pragma unroll
  for (int j = 0; j < 4; ++j) {
    const int col = colbase + 16 * j + l16;
    const float bv = bias[col];
    #pragma unroll
    for (int r = 0; r < 8; ++r) {
      const int row = rowbase + wave * 16 + r + ((lane >> 4) << 3);
      float x = acc[j][r] + bv;
      if (ACT) x = 0.5f * x * (1.f + erff(x * 0.70710678118654752f)); // exact GELU
      C[(size_t)row * 768 + col] = x;
    }
  }
}

// -------------------------- kernel 5: layernorm ----------------------------
__global__ __launch_bounds__(256)
void k5_layernorm(const float* __restrict__ fused, float* __restrict__ out) {
  __shared__ float red1[8];
  __shared__ float red2[8];
  const int row = blockIdx.x, tid = threadIdx.x;
  const int wave = tid >> 5, lane = tid & 31;
  float x[3];
  float s = 0.f;
  #pragma unroll
  for (int j = 0; j < 3; ++j) {
    x[j] = fused[(size_t)row * 768 + tid + 256 * j];
    s += x[j];
  }
  #pragma unroll
  for (int off = 16; off; off >>= 1) s += __shfl_xor(s, off, 32);
  if (lane == 0) red1[wave] = s;
  __syncthreads();
  float tot = 0.f;
  #pragma unroll
  for (int w = 0; w < 8; ++w) tot += red1[w];
  const float mu = tot * (1.f / 768.f);

  float vs = 0.f;
  #pragma unroll
  for (int j = 0; j < 3; ++j) { float d = x[j] - mu; vs += d * d; }
  #pragma unroll
  for (int off = 16; off; off >>= 1) vs += __shfl_xor(vs, off, 32);
  if (lane == 0) red2[wave] = vs;
  __syncthreads();
  float vtot = 0.f;
  #pragma unroll
  for (int w = 0; w < 8; ++w) vtot += red2[w];
  const float rs = rsqrtf(vtot * (1.f / 768.f) + 1e-5f);
  #pragma unroll
  for (int j = 0; j < 3; ++j)
    out[(size_t)row * 768 + tid + 256 * j] = (x[j] - mu) * rs;
}

// ------------------------------- launcher ----------------------------------
extern "C" void kernel_launch(void* const* d_in, const int* in_sizes, int n_in,
                              void* d_out, int out_size, void* d_ws, size_t ws_size,
                              hipStream_t stream) {
  (void)in_sizes; (void)n_in; (void)out_size; (void)ws_size;
  const float* vseq = (const float*)d_in[0];
  const float* Tc   = (const float*)d_in[1];
  const float* nb   = (const float*)d_in[2];
  const float* W1   = (const float*)d_in[3];
  const float* b1   = (const float*)d_in[4];
  const float* W2   = (const float*)d_in[5];
  const float* b2   = (const float*)d_in[6];
  float* out = (float*)d_out;

  float* ws = (float*)d_ws;
  float* pval     = ws;                                  // 25*2048*3
  int*   pidx     = (int*)(ws + NCHUNK * ROWS * 3);      // 25*2048*3
  float* combined = ws + 2 * NCHUNK * ROWS * 3;          // 2048*1680
  float* h        = combined + (size_t)ROWS * CT;        // 2048*768
  float* fused    = h + (size_t)ROWS * 768;              // 2048*768

  const size_t smem1 = (size_t)(K1_BLOCK_M + 16) * APAD * sizeof(float); // 247 KB
  k1_sims_topk<<<dim3(ROWS / K1_BLOCK_M, NCHUNK), K1_THREADS, smem1, stream>>>(
      vseq, Tc, pval, pidx);

  k2_merge_build<<<dim3(ROWS / 4), 128, 0, stream>>>(vseq, nb, pval, pidx, combined);

  gemm_kernel<120, 124, CT, TOTAL_IN, 1>
      <<<dim3(ROWS / 64, 768 / 64), 128, (size_t)2 * 64 * 124 * sizeof(float), stream>>>(
          combined, CT, W1, b1, h);

  gemm_kernel<128, 132, 768, 768, 0>
      <<<dim3(ROWS / 64, 768 / 64), 128, (size_t)2 * 64 * 132 * sizeof(float), stream>>>(
          h, 768, W2, b2, fused);

  k5_layernorm<<<dim3(ROWS), 256, 0, stream>>>(fused, out);
}